// ROILabel_55722905698380
// MI455X (gfx1250) — compile-verified
//
#include <hip/hip_runtime.h>
#include <stdint.h>

#define NUM_ROIS    8192
#define NUM_CLASSES 20
#define NUM_CHUNKS  32            // 8192 / 256
#define FG_THRESH   0.5f
#define BG_HI       0.5f
#define BG_LO       0.1f

// ---------------------------------------------------------------------------
// Kernel 1: single coalesced pass over S ([8192 x 20] f32, row-major).
// 32 blocks x 256 threads, one row per thread (5x float4 = global_load_b128).
// Per-class block reduction: LDS transpose -> per-wave scan + wave32 shfl tree
// with first-index tie-breaking (jnp.argmax semantics). Emits 32 partial
// (value, index) pairs per class into the workspace.
// ---------------------------------------------------------------------------
__global__ __launch_bounds__(256) void seed_partial_kernel(
    const float* __restrict__ S,
    float* __restrict__ pval,
    int*   __restrict__ pidx) {
  __shared__ float sval[NUM_CLASSES * 256];

  const int t = threadIdx.x;
  const int n = blockIdx.x * 256 + t;

  // One row per thread: 20 floats = 5 x float4, 16B-aligned (80B rows).
  const float4* row = (const float4*)(S + (size_t)n * NUM_CLASSES);
  float r[NUM_CLASSES];
  {
    float4 q;
    q = row[0]; r[0] = q.x;  r[1] = q.y;  r[2] = q.z;  r[3] = q.w;
    q = row[1]; r[4] = q.x;  r[5] = q.y;  r[6] = q.z;  r[7] = q.w;
    q = row[2]; r[8] = q.x;  r[9] = q.y;  r[10] = q.z; r[11] = q.w;
    q = row[3]; r[12] = q.x; r[13] = q.y; r[14] = q.z; r[15] = q.w;
    q = row[4]; r[16] = q.x; r[17] = q.y; r[18] = q.z; r[19] = q.w;
  }
#pragma unroll
  for (int c = 0; c < NUM_CLASSES; ++c) sval[c * 256 + t] = r[c];
  __syncthreads();

  const int wid  = t >> 5;   // wave id (8 waves of 32 on CDNA5)
  const int lane = t & 31;

  for (int c = wid; c < NUM_CLASSES; c += 8) {
    float best = -1.0f;           // S values are in [0,1)
    int   bidx = 0x7fffffff;
#pragma unroll
    for (int k = 0; k < 8; ++k) {
      const int tt  = k * 32 + lane;                 // conflict-free LDS
      const float v = sval[c * 256 + tt];
      const int  idx = blockIdx.x * 256 + tt;
      if (v > best || (v == best && idx < bidx)) { best = v; bidx = idx; }
    }
    // wave32 tree reduction with first-index tie-break
#pragma unroll
    for (int off = 16; off > 0; off >>= 1) {
      const float v = __shfl_down(best, off, 32);
      const int   i = __shfl_down(bidx, off, 32);
      if (v > best || (v == best && i < bidx)) { best = v; bidx = i; }
    }
    if (lane == 0) {
      pval[c * NUM_CHUNKS + blockIdx.x] = best;
      pidx[c * NUM_CHUNKS + blockIdx.x] = bidx;
    }
  }
}

// ---------------------------------------------------------------------------
// Kernel 2: ROI labeling.
// Step A: each block redundantly finishes the seed argmax from the 20x32
//         partials (640 L2-hot entries; cheaper than a third launch).
// Step B: exploit symmetry of U (U[n,seed] == U[seed,n]) so the per-class
//         gather becomes contiguous row reads, staged to LDS with CDNA5
//         async global->LDS copies (ASYNCcnt), drained by s_wait_asynccnt.
// Step C: max / first-index argmax over valid classes + fg/bg thresholds.
// ---------------------------------------------------------------------------
__global__ __launch_bounds__(256) void label_kernel(
    const float* __restrict__ U,
    const float* __restrict__ L,
    const float* __restrict__ CW,
    const float* __restrict__ pval,
    const int*   __restrict__ pidx,
    int*         __restrict__ RL,
    float*       __restrict__ RW) {
  __shared__ float tile[NUM_CLASSES * 256];
  __shared__ int   sseed[NUM_CLASSES];
  __shared__ float sw[NUM_CLASSES];
  __shared__ int   svalid[NUM_CLASSES];

  const int t = threadIdx.x;
  const int n = blockIdx.x * 256 + t;

  if (t < NUM_CLASSES) {
    float best = -1.0f;
    int   bidx = 0x7fffffff;
    for (int b = 0; b < NUM_CHUNKS; ++b) {       // ascending chunk = ascending idx
      const float v = pval[t * NUM_CHUNKS + b];
      const int   i = pidx[t * NUM_CHUNKS + b];
      if (v > best || (v == best && i < bidx)) { best = v; bidx = i; }
    }
    sseed[t]  = bidx;
    sw[t]     = CW[t];
    svalid[t] = (L[t] > 0.0f) ? 1 : 0;
  }
  __syncthreads();

  // Async stage: tile[c*256 + t] = U[seed_c][blockStart + t] (coalesced rows).
  for (int c = 0; c < NUM_CLASSES; ++c) {
    if (svalid[c]) {                              // uniform branch per block
      const float* gp = U + (size_t)sseed[c] * (size_t)NUM_ROIS + (size_t)n;
      unsigned lds_off =
          (unsigned)(uintptr_t)(const void*)&tile[c * 256 + t];  // low 32b = LDS offset
      asm volatile("global_load_async_to_lds_b32 %0, %1, off"
                   :
                   : "v"(lds_off), "v"(gp)
                   : "memory");
    }
  }
  asm volatile("s_wait_asynccnt 0x0" ::: "memory");
  __syncthreads();

  // Max / first-index argmax over valid classes (invalid == -1.0, index 0).
  float best  = -1.0f;
  int   bestc = 0;
  for (int c = 0; c < NUM_CLASSES; ++c) {
    if (svalid[c]) {
      const float v = tile[c * 256 + t];
      if (v > best) { best = v; bestc = c; }
    }
  }

  const bool fg = (best >= FG_THRESH);
  const bool bg = (best < BG_HI) && (best >= BG_LO);
  RL[n] = fg ? (bestc + 1) : (bg ? 0 : -1);
  RW[n] = (fg || bg) ? sw[bestc] : 0.0f;
}

// ---------------------------------------------------------------------------
// Launcher. Inputs (setup_inputs order): S [8192x20] f32, U [8192x8192] f32,
// L [1x20] f32, CW [1x20] f32. Output: RL [8192] int32 then RW [8192] f32,
// concatenated flat in d_out. Workspace: 640 partial values + 640 indices.
// ---------------------------------------------------------------------------
extern "C" void kernel_launch(void* const* d_in, const int* in_sizes, int n_in,
                              void* d_out, int out_size, void* d_ws, size_t ws_size,
                              hipStream_t stream) {
  const float* S  = (const float*)d_in[0];
  const float* U  = (const float*)d_in[1];
  const float* L  = (const float*)d_in[2];
  const float* CW = (const float*)d_in[3];

  float* pval = (float*)d_ws;                               // 20*32 floats
  int*   pidx = (int*)((float*)d_ws + NUM_CLASSES * NUM_CHUNKS);  // 20*32 ints

  int*   RL = (int*)d_out;
  float* RW = (float*)d_out + NUM_ROIS;

  seed_partial_kernel<<<NUM_CHUNKS, 256, 0, stream>>>(S, pval, pidx);
  label_kernel<<<NUM_ROIS / 256, 256, 0, stream>>>(U, L, CW, pval, pidx, RL, RW);
}